// MultiScaleDeformableAttention_66305705116354
// MI455X (gfx1250) — compile-verified
//
#include <hip/hip_runtime.h>

// ---------------------------------------------------------------------------
// Multi-Scale Deformable Attention for MI455X (gfx1250, wave32, WMMA)
//
// DIM=256, H=8, L=4, P=4, HEAD_DIM=32, B=8, Q=1024, S_TOTAL=12240.
// All GEMMs have K=256; problem is HBM-bandwidth-bound (~210MB moved,
// ~15.5 GFLOP), so full-precision V_WMMA_F32_16X16X4_F32 is the right path.
// Tile staging uses gfx1250 GLOBAL_LOAD_ASYNC_TO_LDS_B128 (ASYNCcnt) when
// the builtin is available, with a register-staged fallback.
// ---------------------------------------------------------------------------

typedef __attribute__((ext_vector_type(2))) float v2f;
typedef __attribute__((ext_vector_type(8))) float v8f;
typedef __attribute__((ext_vector_type(4))) int   v4i_t;
typedef __attribute__((address_space(1))) v4i_t   v4i_glb;   // global (__device__)
typedef __attribute__((address_space(3))) v4i_t   v4i_lds;   // LDS

#if defined(__has_builtin)
#if __has_builtin(__builtin_amdgcn_global_load_async_to_lds_b128) && \
    __has_builtin(__builtin_amdgcn_s_wait_asynccnt)
#define USE_ASYNC_LDS 1
#endif
#endif

#define LDA_S 132   // A tile [32 m][128 k] + 4 skew: 528B rows (16B aligned, 4-bank skew)
#define LDB_S 40    // B tile [128 k][32 n] + 8 skew: half-wave k+2 rows -> +16 banks (disjoint)

#ifdef USE_ASYNC_LDS
__device__ __forceinline__ void cp16_async(const float* g, float* l) {
    // per-lane 16B global->LDS copy, tracked by ASYNCcnt (no VGPR round-trip).
    // Global ptrs are 64-bit AS1; LDS ptrs are 32-bit AS3 (low 32 bits of the
    // generic pointer to a __shared__ object are the AS3 offset on AMDGCN).
    __builtin_amdgcn_global_load_async_to_lds_b128(
        (v4i_glb*)(__UINTPTR_TYPE__)g,
        (v4i_lds*)(unsigned int)(__UINTPTR_TYPE__)l,
        /*offset=*/0, /*cpol=*/0);
}
#endif

// C[M,N] = A[M,256] * B[256,N] + bias[N]
// 128 threads = 4 waves; 2x2 waves of 16x16 WMMA tiles -> 32x32 block tile.
__global__ __launch_bounds__(128)
void gemm256_wmma_f32(const float* __restrict__ A, const float* __restrict__ Bm,
                      const float* __restrict__ bias, float* __restrict__ C,
                      int N) {
    __shared__ float sA[32 * LDA_S];    // [m][k] chunk, row-major
    __shared__ float sB[128 * LDB_S];   // [k][n] chunk, row-major

    const int tid  = threadIdx.x;
    const int lane = tid & 31;
    const int wave = tid >> 5;
    const int wx   = wave & 1;
    const int wy   = wave >> 1;
    const int n0   = blockIdx.x * 32;
    const int m0   = blockIdx.y * 32;
    const int l16  = lane & 15;        // row (A) / col (B) within fragment
    const int lh   = lane >> 4;        // lanes 0-15 -> K{4k,4k+1}; 16-31 -> K{4k+2,4k+3}

    v8f acc = {};

    for (int kc = 0; kc < 2; ++kc) {   // two K-chunks of 128
#ifdef USE_ASYNC_LDS
        // ---- async tile staging: 16 in-flight 16B/lane copies, one wait ----
        #pragma unroll
        for (int i = 0; i < 8; ++i) {
            int f = tid + 128 * i;              // A: 1024 float4 units
            int row = f >> 5, c4 = f & 31;
            cp16_async(A + (size_t)(m0 + row) * 256 + kc * 128 + c4 * 4,
                       sA + row * LDA_S + c4 * 4);
        }
        #pragma unroll
        for (int i = 0; i < 8; ++i) {
            int f = tid + 128 * i;              // B: 1024 float4 units
            int k = f >> 3, c4 = f & 7;
            cp16_async(Bm + (size_t)(kc * 128 + k) * N + n0 + c4 * 4,
                       sB + k * LDB_S + c4 * 4);
        }
        __builtin_amdgcn_s_wait_asynccnt(0);
#else
        // ---- fallback: stage through registers, all loads in flight first ----
        float4 ra[8], rb[8];
        #pragma unroll
        for (int i = 0; i < 8; ++i) {
            int f = tid + 128 * i;
            int row = f >> 5, c4 = f & 31;
            ra[i] = *(const float4*)(A + (size_t)(m0 + row) * 256 + kc * 128 + c4 * 4);
        }
        #pragma unroll
        for (int i = 0; i < 8; ++i) {
            int f = tid + 128 * i;
            int k = f >> 3, c4 = f & 7;
            rb[i] = *(const float4*)(Bm + (size_t)(kc * 128 + k) * N + n0 + c4 * 4);
        }
        #pragma unroll
        for (int i = 0; i < 8; ++i) {
            int f = tid + 128 * i;
            int row = f >> 5, c4 = f & 31;
            *(float4*)(sA + row * LDA_S + c4 * 4) = ra[i];
        }
        #pragma unroll
        for (int i = 0; i < 8; ++i) {
            int f = tid + 128 * i;
            int k = f >> 3, c4 = f & 7;
            *(float4*)(sB + k * LDB_S + c4 * 4) = rb[i];
        }
#endif
        __syncthreads();

        // ISA 16x4 f32 A fragment: lane<16 holds K{4k,4k+1}, lane>=16 K{4k+2,4k+3}
        const float* aBase = sA + (wy * 16 + l16) * LDA_S + lh * 2;
        // 4x16 B fragment: same K selection, col = l16 (b.x/b.y one k-row apart)
        const float* bBase = sB + (lh * 2) * LDB_S + wx * 16 + l16;

        #pragma unroll
        for (int k = 0; k < 32; ++k) {
            v2f a = *(const v2f*)(aBase + 4 * k);       // ds_load_b64
            v2f b;
            b.x = bBase[(4 * k + 0) * LDB_S];           // merge -> ds_load_2addr_b32
            b.y = bBase[(4 * k + 1) * LDB_S];
            acc = __builtin_amdgcn_wmma_f32_16x16x4_f32(
                /*neg_a=*/false, a, /*neg_b=*/false, b,
                /*c_mod=*/(short)0, acc, /*reuse_a=*/false, /*reuse_b=*/false);
        }
        __syncthreads();
    }

    // C/D layout: VGPR j holds row (j + lh*8), col l16 within the 16x16 tile
    const int n  = n0 + wx * 16 + l16;
    const float bv = bias ? bias[n] : 0.0f;
    const int rbase = m0 + wy * 16 + lh * 8;
    #pragma unroll
    for (int j = 0; j < 8; ++j)
        C[(size_t)(rbase + j) * N + n] = acc[j] + bv;
}

// ---------------------------------------------------------------------------
// Sampler: one wave per (b*q, head); lane = channel d in [0,32).
// ---------------------------------------------------------------------------
__global__ __launch_bounds__(256)
void msda_sample(const float* __restrict__ attn_raw,   // [8192][128]
                 const float* __restrict__ offs,       // [8192][256]
                 const float* __restrict__ refp,       // [8192][4][2]
                 const float* __restrict__ V,          // [8*12240][256]
                 float* __restrict__ outH) {           // [8192][256]
    const int tid  = threadIdx.x;
    const int lane = tid & 31;
    const int wave = tid >> 5;
    const int unit = blockIdx.x * 8 + wave;   // 0..65535
    const int bq   = unit >> 3;
    const int h    = unit & 7;
    const int b    = bq >> 10;                // Q = 1024

    // softmax over the 16 (level,point) logits; broadcast loads across lanes
    const float* lg = attn_raw + (size_t)bq * 128 + h * 16;
    float w[16];
    float mx = -1e30f;
    #pragma unroll
    for (int i = 0; i < 16; ++i) { w[i] = lg[i]; mx = fmaxf(mx, w[i]); }
    float s = 0.0f;
    #pragma unroll
    for (int i = 0; i < 16; ++i) { w[i] = __expf(w[i] - mx); s += w[i]; }
    const float inv = 1.0f / s;

    const int HH[4] = {96, 48, 24, 12};
    const int ST[4] = {0, 9216, 11520, 12096};

    const float* ofs = offs + (size_t)bq * 256 + h * 32;  // h*L*P*2
    const float* rp  = refp + (size_t)bq * 8;             // L*2

    float acc = 0.0f;
    #pragma unroll
    for (int lvl = 0; lvl < 4; ++lvl) {
        const int hh = HH[lvl];
        const int ww = HH[lvl];                 // square levels
        const float r0 = rp[lvl * 2 + 0];
        const float r1 = rp[lvl * 2 + 1];
        #pragma unroll
        for (int p = 0; p < 4; ++p) {
            float o0 = ofs[(lvl * 4 + p) * 2 + 0];
            float o1 = ofs[(lvl * 4 + p) * 2 + 1];
            float l0 = fminf(fmaxf(r0 + o0 * 0.1f, 0.0f), 1.0f);
            float l1 = fminf(fmaxf(r1 + o1 * 0.1f, 0.0f), 1.0f);
            int y = (int)(l0 * (float)(hh - 1)); y = min(max(y, 0), hh - 1);
            int x = (int)(l1 * (float)(ww - 1)); x = min(max(x, 0), ww - 1);
            size_t row = (size_t)b * 12240 + ST[lvl] + y * ww + x;
            acc += (w[lvl * 4 + p] * inv) * V[row * 256 + h * 32 + lane];
        }
    }
    outH[(size_t)bq * 256 + h * 32 + lane] = acc;
}

// ---------------------------------------------------------------------------
// Launch: V = input_flatten@W_val+b_val ; OFFS/ATTN = query proj ; sample ;
//         out = OUTH@W_out + b_out
// ---------------------------------------------------------------------------
extern "C" void kernel_launch(void* const* d_in, const int* in_sizes, int n_in,
                              void* d_out, int out_size, void* d_ws, size_t ws_size,
                              hipStream_t stream) {
    (void)in_sizes; (void)n_in; (void)out_size; (void)ws_size;

    const float* query  = (const float*)d_in[0];
    const float* refp   = (const float*)d_in[1];
    const float* inpf   = (const float*)d_in[2];
    // d_in[3] spatial shapes: compile-time constants {96,48,24,12}
    const float* W_off  = (const float*)d_in[4];
    const float* b_off  = (const float*)d_in[5];
    const float* W_attn = (const float*)d_in[6];
    const float* b_attn = (const float*)d_in[7];
    const float* W_val  = (const float*)d_in[8];
    const float* b_val  = (const float*)d_in[9];
    const float* W_out  = (const float*)d_in[10];
    const float* b_out  = (const float*)d_in[11];
    float* out = (float*)d_out;

    float* ws    = (float*)d_ws;
    float* V     = ws;                      // 8*12240*256      = 25,067,520 f
    float* OFFS  = V + 25067520ull;         // 8192*256         =  2,097,152 f
    float* ATTNR = OFFS + 2097152ull;       // 8192*128         =  1,048,576 f
    float* OUTH  = ATTNR + 1048576ull;      // 8192*256         =  2,097,152 f

    const dim3 blk(128);
    // value projection: M = 97920, N = 256
    gemm256_wmma_f32<<<dim3(256 / 32, 97920 / 32), blk, 0, stream>>>(inpf, W_val, b_val, V, 256);
    // offset projection: M = 8192, N = 256
    gemm256_wmma_f32<<<dim3(256 / 32, 8192 / 32), blk, 0, stream>>>(query, W_off, b_off, OFFS, 256);
    // attention-logit projection: M = 8192, N = 128
    gemm256_wmma_f32<<<dim3(128 / 32, 8192 / 32), blk, 0, stream>>>(query, W_attn, b_attn, ATTNR, 128);
    // softmax + deformable gather + weighted sum
    msda_sample<<<dim3(8192), dim3(256), 0, stream>>>(ATTNR, OFFS, refp, V, OUTH);
    // output projection: M = 8192, N = 256 -> d_out
    gemm256_wmma_f32<<<dim3(256 / 32, 8192 / 32), blk, 0, stream>>>(OUTH, W_out, b_out, out, 256);
}